// GCNLayer_31499290149286
// MI455X (gfx1250) — compile-verified
//
#include <hip/hip_runtime.h>
#include <cstdint>
#include <cstddef>

// Problem constants (match reference).
#define D_FEAT_C   64
#define N_NODES_C  100000

#define EDGES_PER_BLOCK  64
#define THREADS_PER_EDGE 16   // 16 lanes x float4 = 64 floats per edge row
#define EDGES_PER_GROUP  4    // each 16-lane group loops over 4 edges

// ---- gfx1250 async-to-LDS feature detection -------------------------------
#if defined(__has_builtin)
#  if __has_builtin(__builtin_amdgcn_global_load_async_to_lds_b128)
#    define HAVE_ASYNC_B128 1
#  endif
#  if __has_builtin(__builtin_amdgcn_s_wait_asynccnt)
#    define HAVE_WAIT_ASYNC 1
#  endif
#endif

// Builtin operand types: AS1 (global) and AS3 (LDS) pointers to a 4 x i32
// vector (signature confirmed by clang diagnostics).
typedef int v4i __attribute__((ext_vector_type(4)));
typedef __attribute__((address_space(1))) v4i g1_v4i;
typedef __attribute__((address_space(3))) v4i l3_v4i;

// Generic (flat) pointer -> AS-qualified pointer. For LDS, the low 32 bits of
// a flat LDS address are the wave-relative LDS byte offset (ISA 10.2 aperture
// mapping); for global memory the flat address equals the global address.
__device__ __forceinline__ g1_v4i* to_global_v4(const void* p) {
    return (g1_v4i*)(uintptr_t)p;
}
__device__ __forceinline__ l3_v4i* to_lds_v4(void* p) {
    return (l3_v4i*)(unsigned)(uintptr_t)p;
}

// ---------------------------------------------------------------------------
// Kernel 1: zero the output accumulator and the per-node counts.
// (Harness poisons d_out/d_ws with 0xAA; must be idempotent per launch.)
__global__ void gcn_zero(float* __restrict__ out, float* __restrict__ counts) {
    unsigned i = blockIdx.x * blockDim.x + threadIdx.x;
    const unsigned total = N_NODES_C * D_FEAT_C;
    if (i < total) out[i] = 0.0f;
    if (i < N_NODES_C) counts[i] = 0.0f;
}

// ---------------------------------------------------------------------------
// Kernel 2: edge scatter.  64 edges per 256-thread block.  The 128 edge
// indices (512 B) are staged into LDS by 32 lanes of async DMA
// (global_load_async_to_lds_b128, ASYNCcnt).  Each 16-lane group then loops
// over 4 edges: one 256B row gather (global_load_b128 per lane, L2-hot since
// the 25.6MB table fits the 192MB L2) and 4 no-return global_atomic_add_f32
// per lane into the L2-resident accumulator.  Unrolled loop keeps 4 row
// fetches + 16 atomics in flight per thread for latency hiding.
__global__ __launch_bounds__(256) void gcn_scatter(
    const float* __restrict__ emb,
    const int*   __restrict__ src,
    const int*   __restrict__ dst,
    float*       __restrict__ out,
    float*       __restrict__ counts)
{
    __shared__ int s_idx[2 * EDGES_PER_BLOCK];  // [0..63]=src, [64..127]=dst

    const unsigned t     = threadIdx.x;
    const unsigned ebase = blockIdx.x * EDGES_PER_BLOCK;

#if defined(HAVE_ASYNC_B128)
    // 32 lanes x b128 = 512 B of indices, one async DMA batch from wave 0.
    if (t < 32) {
        const int* gp = (t < 16) ? (src + ebase + t * 4)
                                 : (dst + ebase + (t - 16) * 4);
        __builtin_amdgcn_global_load_async_to_lds_b128(
            to_global_v4(gp), to_lds_v4(&s_idx[t * 4]), /*offset=*/0, /*cpol=*/0);
    }
#  if defined(HAVE_WAIT_ASYNC)
    __builtin_amdgcn_s_wait_asynccnt(0);
#  else
    asm volatile("s_wait_asynccnt 0" ::: "memory");
#  endif
    __syncthreads();
#else
    if (t < 2 * EDGES_PER_BLOCK) {
        s_idx[t] = (t < EDGES_PER_BLOCK) ? src[ebase + t]
                                         : dst[ebase + t - EDGES_PER_BLOCK];
    }
    __syncthreads();
#endif

    const unsigned g = t / THREADS_PER_EDGE;   // group 0..15
    const unsigned l = t % THREADS_PER_EDGE;   // float4 lane within edge

#pragma unroll
    for (unsigned k = 0; k < EDGES_PER_GROUP; ++k) {
        const unsigned e_local = g * EDGES_PER_GROUP + k;   // 0..63
        const int sn = s_idx[e_local];
        const int dn = s_idx[EDGES_PER_BLOCK + e_local];

        const float4 v = *((const float4*)(emb + (size_t)sn * D_FEAT_C) + l);

        float* o = out + (size_t)dn * D_FEAT_C + l * 4;
        atomicAdd(o + 0, v.x);
        atomicAdd(o + 1, v.y);
        atomicAdd(o + 2, v.z);
        atomicAdd(o + 3, v.w);

        if (l == 0) atomicAdd(counts + dn, 1.0f);
    }
}

// Scalar tail for edge counts not divisible by EDGES_PER_BLOCK.
__global__ void gcn_scatter_tail(
    const float* __restrict__ emb,
    const int*   __restrict__ src,
    const int*   __restrict__ dst,
    float*       __restrict__ out,
    float*       __restrict__ counts,
    int start, int n_edges)
{
    const int t = blockIdx.x * blockDim.x + threadIdx.x;
    const int e = start + t / THREADS_PER_EDGE;
    const int l = t % THREADS_PER_EDGE;
    if (e >= n_edges) return;

    const int sn = src[e];
    const int dn = dst[e];
    const float4 v = *((const float4*)(emb + (size_t)sn * D_FEAT_C) + l);
    float* o = out + (size_t)dn * D_FEAT_C + l * 4;
    atomicAdd(o + 0, v.x);
    atomicAdd(o + 1, v.y);
    atomicAdd(o + 2, v.z);
    atomicAdd(o + 3, v.w);
    if (l == 0) atomicAdd(counts + dn, 1.0f);
}

// ---------------------------------------------------------------------------
// Kernel 3: mean normalize, float4 per thread, IEEE division to match the
// reference's summed / max(counts, 1).
__global__ void gcn_normalize(float* __restrict__ out,
                              const float* __restrict__ counts) {
    const unsigned i = blockIdx.x * blockDim.x + threadIdx.x;
    const unsigned total4 = (N_NODES_C * D_FEAT_C) / 4;
    if (i >= total4) return;

    const unsigned node = i / (D_FEAT_C / 4);
    const float denom = fmaxf(counts[node], 1.0f);

    float4* p = (float4*)out + i;
    float4 v = *p;
    v.x /= denom;
    v.y /= denom;
    v.z /= denom;
    v.w /= denom;
    *p = v;
}

// ---------------------------------------------------------------------------
extern "C" void kernel_launch(void* const* d_in, const int* in_sizes, int n_in,
                              void* d_out, int out_size, void* d_ws, size_t ws_size,
                              hipStream_t stream) {
    const float* emb = (const float*)d_in[0];   // [N_NODES, 64] f32
    const int*   src = (const int*)d_in[1];     // [E] i32
    const int*   dst = (const int*)d_in[2];     // [E] i32
    float* out    = (float*)d_out;              // [N_NODES, 64] f32
    float* counts = (float*)d_ws;               // N_NODES f32 scratch

    const int n_edges = in_sizes[1];
    const int total   = N_NODES_C * D_FEAT_C;

    gcn_zero<<<(total + 255) / 256, 256, 0, stream>>>(out, counts);

    const int full_blocks = n_edges / EDGES_PER_BLOCK;
    if (full_blocks > 0) {
        gcn_scatter<<<full_blocks, 256, 0, stream>>>(emb, src, dst, out, counts);
    }
    const int rem = n_edges % EDGES_PER_BLOCK;
    if (rem > 0) {
        gcn_scatter_tail<<<(rem * THREADS_PER_EDGE + 255) / 256, 256, 0, stream>>>(
            emb, src, dst, out, counts,
            full_blocks * EDGES_PER_BLOCK, n_edges);
    }

    gcn_normalize<<<(total / 4 + 255) / 256, 256, 0, stream>>>(out, counts);
}